// GVPEncoder_21706764714038
// MI455X (gfx1250) — compile-verified
//
#include <hip/hip_runtime.h>
#include <math.h>

// ---------------------------------------------------------------------------
// GVP-GNN encoder for MI455X (gfx1250, wave32).
// Heavy GEMMs run on the matrix pipe via V_WMMA_F32_16X16X4_F32 (fp32 in/out,
// exact fp32 accumulate -> matches the fp32 reference numerically).
// Edge gathers (s[src]|es|s[dst]|vn) are fused into the GEMM's LDS A-staging.
// Software-pipelined: global loads for chunk i+1 prefetched to registers while
// chunk i computes; B LDS double-buffered (A tile is wave-local -> no barrier
// needed for it); one s_barrier per 32-wide K chunk.
// ---------------------------------------------------------------------------

typedef float v2f __attribute__((ext_vector_type(2)));
typedef float v8f __attribute__((ext_vector_type(8)));

#define NN 100000      // nodes
#define NE 400000      // edges
#define NG 2000        // graphs
#define SD 256         // scalar dim
#define EPSQ 1e-8f
#define LNEPS 1e-5f

#define TM 128
#define TN 128
#define TKB 32
#define TNP 144        // pair-row stride (float2 units); 288 dwords == 32 mod 64
                       // -> the two half-waves (rows p, p+1) hit disjoint banks

// ===========================================================================
// Tiled GEMM with gathered/concatenated A rows:
//   C[m, n] = act( sum_k A_row(m)[k] * W[k, n] + bias[n] )
// A_row(m) = concat of up to 4 segments; each segment i has row length L_i and
// row index selected by S_i: 0 -> m, 1 -> src[m], 2 -> dst[m].
// WMMA fragment layouts per CDNA5 ISA 7.12.2:
//   A 16x4 f32: lanes 0-15 hold M=lane, {v0,v1}={K+0,K+1}; lanes 16-31 {K+2,K+3}
//   B assumed symmetric; C/D: vgpr r -> M=r (lanes<16) / r+8 (lanes>=16), N=lane&15
// ===========================================================================
__global__ __launch_bounds__(256) void gemm_gather_wmma(
    const float* __restrict__ A0, int L0, int S0,
    const float* __restrict__ A1, int L1, int S1,
    const float* __restrict__ A2, int L2, int S2,
    const float* __restrict__ A3, int L3, int S3,
    const int* __restrict__ srcI, const int* __restrict__ dstI,
    const float* __restrict__ W, const float* __restrict__ bias,
    float* __restrict__ C, int M, int K, int N, int act)
{
    // A tile is wave-local (each wave writes and reads the same 16 rows), so a
    // single buffer suffices (per-wave LDS ordering). B is block-shared ->
    // double-buffered for the pipelined loop.
    __shared__ __align__(16) float As[TM][TKB + 4];
    __shared__ v2f Bs2[2][(TKB / 2) * TNP];  // Bs2[buf][p*TNP+n] = {W[2p][n],W[2p+1][n]}

    const int tid  = threadIdx.x;
    const int lane = tid & 31;
    const int wid  = tid >> 5;
    const int m0   = blockIdx.x * TM;
    const int n0   = blockIdx.y * TN;

    v8f acc[TN / 16] = {};

    // A-load mapping: thread -> one row, 16 consecutive k (rows are wave-local)
    const int arow_l = tid >> 1;
    const int akoff  = (tid & 1) * 16;
    const int am     = m0 + arow_l;
    const bool needS = (S0 == 1) | (S1 == 1) | (S2 == 1) | (S3 == 1);
    const bool needD = (S0 == 2) | (S1 == 2) | (S2 == 2) | (S3 == 2);
    int asrc = 0, adst = 0;
    if (am < M) {
        if (needS) asrc = srcI[am];
        if (needD) adst = dstI[am];
    }
    // B-load mapping: thread -> one k row, 16 consecutive n
    const int bk_l  = tid >> 3;
    const int bnoff = (tid & 7) * 16;

    const int rlocal = wid * 16;          // this wave's 16 rows
    const int l15    = lane & 15;
    const int koff   = (lane >> 4) * 2;

    float aReg[16];
    float bReg[16];

    auto fetchA = [&](int kb) {
        #pragma unroll
        for (int j = 0; j < 16; ++j) {
            int kg = kb + akoff + j;
            float val = 0.0f;
            if (am < M && kg < K) {
                int kk = kg;
                const float* base; int len; int sel;
                if (kk < L0)               { base = A0; len = L0; sel = S0; }
                else if ((kk -= L0) < L1)  { base = A1; len = L1; sel = S1; }
                else if ((kk -= L1) < L2)  { base = A2; len = L2; sel = S2; }
                else { kk -= L2;             base = A3; len = L3; sel = S3; }
                int r = (sel == 0) ? am : ((sel == 1) ? asrc : adst);
                val = base[(size_t)r * len + kk];
            }
            aReg[j] = val;
        }
    };
    auto storeA = [&]() {
        #pragma unroll
        for (int j = 0; j < 16; ++j) As[arow_l][akoff + j] = aReg[j];
    };
    auto fetchB = [&](int kb) {
        int kg = kb + bk_l;
        #pragma unroll
        for (int j = 0; j < 16; ++j) {
            int n = n0 + bnoff + j;
            bReg[j] = (kg < K && n < N) ? W[(size_t)kg * N + n] : 0.0f;
        }
    };
    auto storeB = [&](int buf) {
        float* BsF = (float*)&Bs2[buf][0];
        int prow = (bk_l >> 1) * TNP;
        int par  = bk_l & 1;
        #pragma unroll
        for (int j = 0; j < 16; ++j)
            BsF[(prow + bnoff + j) * 2 + par] = bReg[j];
    };

    const int nkb = (K + TKB - 1) / TKB;

    // prologue: stage chunk 0
    fetchA(0);
    fetchB(0);
    storeA();
    storeB(0);
    __syncthreads();

    for (int ib = 0; ib < nkb; ++ib) {
        const int  cur  = ib & 1;
        const bool more = (ib + 1) < nkb;
        // issue global prefetch for next chunk; latency hidden by the WMMAs
        if (more) { fetchA((ib + 1) * TKB); fetchB((ib + 1) * TKB); }
        // ---- WMMA over current chunk ----
        const int arow = rlocal + l15;
        #pragma unroll
        for (int kk = 0; kk < TKB; kk += 4) {
            v2f a = *(const v2f*)&As[arow][kk + koff];
            const v2f* brow = &Bs2[cur][((kk + koff) >> 1) * TNP + l15];
            v2f b[TN / 16];
            #pragma unroll
            for (int t = 0; t < TN / 16; ++t) b[t] = brow[t * 16];
            #pragma unroll
            for (int t = 0; t < TN / 16; ++t)
                acc[t] = __builtin_amdgcn_wmma_f32_16x16x4_f32(
                    false, a, false, b[t], (short)0, acc[t], false, false);
        }
        // ---- stage next chunk ----
        if (more) {
            storeA();          // wave-local: per-wave LDS ordering protects reads above
            storeB(cur ^ 1);   // other buffer: no conflict with this chunk's readers
        }
        __syncthreads();
    }

    // ---- epilogue: bias + activation ----
    #pragma unroll
    for (int t = 0; t < TN / 16; ++t) {
        union { v8f v; float f[8]; } u;
        u.v = acc[t];
        int col = n0 + t * 16 + l15;
        float bcol = bias ? bias[col] : 0.0f;
        #pragma unroll
        for (int r = 0; r < 8; ++r) {
            int row = m0 + rlocal + r + ((lane >> 4) * 8);
            if (row < M) {
                float x = u.f[r] + bcol;
                if (act == 1) x = fmaxf(x, 0.0f);
                C[(size_t)row * N + col] = x;
            }
        }
    }
}

// ===========================================================================
// GVP vector-channel kernels.
// vn[m,h] = sqrt(max(sum_c (sum_i Vin[m,i,c]*wh[i,h])^2, EPS))
// vout[m,o,c] = sum_i Vin[m,i,c] * P[i,o]   where P = wh @ wv (precomputed)
// Vin rows gathered from up to 3 segments (sel: 0=m, 1=src[m], 2=dst[m]).
// ===========================================================================
__device__ __forceinline__ void accum_seg(
    const float* __restrict__ V, int R, int S, int m,
    const int* __restrict__ srcI, const int* __restrict__ dstI,
    const float* __restrict__ Wm, int Wcols, int colIdx, int& rbase,
    float& a0, float& a1, float& a2)
{
    if (R <= 0 || V == nullptr) return;
    int r = (S == 0) ? m : ((S == 1) ? srcI[m] : dstI[m]);
    const float* p = V + (size_t)r * R * 3;
    for (int i = 0; i < R; ++i) {
        float w = Wm[(rbase + i) * Wcols + colIdx];
        a0 += p[i * 3 + 0] * w;
        a1 += p[i * 3 + 1] * w;
        a2 += p[i * 3 + 2] * w;
    }
    rbase += R;
}

__global__ __launch_bounds__(256) void gvp_vn_kernel(
    const float* V0, int R0, int S0, const float* V1, int R1, int S1,
    const float* V2, int R2, int S2,
    const int* __restrict__ srcI, const int* __restrict__ dstI,
    const float* __restrict__ wh, int H, float* __restrict__ vn, int MH)
{
    int gid = blockIdx.x * blockDim.x + threadIdx.x;
    if (gid >= MH) return;
    int m = gid / H, h = gid - m * H;
    float a0 = 0, a1 = 0, a2 = 0;
    int rb = 0;
    accum_seg(V0, R0, S0, m, srcI, dstI, wh, H, h, rb, a0, a1, a2);
    accum_seg(V1, R1, S1, m, srcI, dstI, wh, H, h, rb, a0, a1, a2);
    accum_seg(V2, R2, S2, m, srcI, dstI, wh, H, h, rb, a0, a1, a2);
    vn[gid] = sqrtf(fmaxf(a0 * a0 + a1 * a1 + a2 * a2, EPSQ));
}

__global__ __launch_bounds__(256) void gvp_vout_kernel(
    const float* V0, int R0, int S0, const float* V1, int R1, int S1,
    const float* V2, int R2, int S2,
    const int* __restrict__ srcI, const int* __restrict__ dstI,
    const float* __restrict__ P, int VO,
    float* __restrict__ vout, int MVO, int act)
{
    int gid = blockIdx.x * blockDim.x + threadIdx.x;
    if (gid >= MVO) return;
    int m = gid / VO, o = gid - m * VO;
    float a0 = 0, a1 = 0, a2 = 0;
    int rb = 0;
    accum_seg(V0, R0, S0, m, srcI, dstI, P, VO, o, rb, a0, a1, a2);
    accum_seg(V1, R1, S1, m, srcI, dstI, P, VO, o, rb, a0, a1, a2);
    accum_seg(V2, R2, S2, m, srcI, dstI, P, VO, o, rb, a0, a1, a2);
    if (act == 1) {  // vout * sigmoid(||vout||)
        float n = sqrtf(fmaxf(a0 * a0 + a1 * a1 + a2 * a2, EPSQ));
        float sg = 1.0f / (1.0f + expf(-n));
        a0 *= sg; a1 *= sg; a2 *= sg;
    }
    vout[(size_t)gid * 3 + 0] = a0;
    vout[(size_t)gid * 3 + 1] = a1;
    vout[(size_t)gid * 3 + 2] = a2;
}

// P = wh (VIxH) @ wv (HxVO)
__global__ void small_matmul_kernel(const float* __restrict__ wh,
                                    const float* __restrict__ wv,
                                    float* __restrict__ P, int VI, int H, int VO)
{
    int id = blockIdx.x * blockDim.x + threadIdx.x;
    if (id >= VI * VO) return;
    int i = id / VO, o = id - i * VO;
    float s = 0.0f;
    for (int h = 0; h < H; ++h) s += wh[i * H + h] * wv[h * VO + o];
    P[id] = s;
}

// ===========================================================================
// LayerNorm (scalar 256-wide) + vector RMS normalization, one wave per node.
// Optional residual: s += adds*scale, v += addv*scale (scale = invcnt or 1).
// ===========================================================================
__device__ __forceinline__ float wave_sum32(float x) {
    #pragma unroll
    for (int i = 16; i > 0; i >>= 1) x += __shfl_xor(x, i, 32);
    return x;
}

__global__ __launch_bounds__(256) void ln_kernel(
    float* __restrict__ s, float* __restrict__ v,
    const float* __restrict__ adds, const float* __restrict__ addv,
    const float* __restrict__ invc,
    const float* __restrict__ gamma, const float* __restrict__ beta, int M)
{
    int node = blockIdx.x * 8 + (threadIdx.x >> 5);
    int lane = threadIdx.x & 31;
    if (node >= M) return;
    float sc = invc ? invc[node] : 1.0f;
    size_t b = (size_t)node * SD;
    float x[8];
    #pragma unroll
    for (int j = 0; j < 8; ++j) {
        int c = j * 32 + lane;
        float t = s[b + c];
        if (adds) t += adds[b + c] * sc;
        x[j] = t;
    }
    float sum = 0;
    #pragma unroll
    for (int j = 0; j < 8; ++j) sum += x[j];
    float mu = wave_sum32(sum) * (1.0f / SD);
    float s2 = 0;
    #pragma unroll
    for (int j = 0; j < 8; ++j) { float d = x[j] - mu; s2 += d * d; }
    float var = wave_sum32(s2) * (1.0f / SD);
    float rstd = 1.0f / sqrtf(var + LNEPS);
    #pragma unroll
    for (int j = 0; j < 8; ++j) {
        int c = j * 32 + lane;
        s[b + c] = (x[j] - mu) * rstd * gamma[c] + beta[c];
    }
    // vector channel: v / sqrt(mean_h max(|v_h|^2, EPS))
    size_t vb = (size_t)node * 48;
    float v0 = 0, v1 = 0, v2 = 0, nh = 0;
    if (lane < 16) {
        v0 = v[vb + lane * 3 + 0];
        v1 = v[vb + lane * 3 + 1];
        v2 = v[vb + lane * 3 + 2];
        if (addv) {
            v0 += addv[vb + lane * 3 + 0] * sc;
            v1 += addv[vb + lane * 3 + 1] * sc;
            v2 += addv[vb + lane * 3 + 2] * sc;
        }
        nh = fmaxf(v0 * v0 + v1 * v1 + v2 * v2, EPSQ);
    }
    float mean = wave_sum32(nh) * (1.0f / 16.0f);
    float rs = 1.0f / sqrtf(mean);
    if (lane < 16) {
        v[vb + lane * 3 + 0] = v0 * rs;
        v[vb + lane * 3 + 1] = v1 * rs;
        v[vb + lane * 3 + 2] = v2 * rs;
    }
}

// ===========================================================================
// Aggregation / pooling helpers
// ===========================================================================
__global__ void count_kernel(const int* __restrict__ idx, float* __restrict__ cnt, int M) {
    int i = blockIdx.x * blockDim.x + threadIdx.x;
    if (i < M) atomicAdd(&cnt[idx[i]], 1.0f);
}
__global__ void scatter_add_kernel(const float* __restrict__ in, const int* __restrict__ idx,
                                   float* __restrict__ out, int M, int D) {
    int gid = blockIdx.x * blockDim.x + threadIdx.x;
    if (gid >= M * D) return;
    int m = gid / D, j = gid - m * D;
    atomicAdd(&out[(size_t)idx[m] * D + j], in[gid]);
}
__global__ void inv_clamp_kernel(float* __restrict__ x, int M) {
    int i = blockIdx.x * blockDim.x + threadIdx.x;
    if (i < M) x[i] = 1.0f / fmaxf(x[i], 1.0f);
}
__global__ void scale_rows_kernel(float* __restrict__ x, const float* __restrict__ inv,
                                  int tot, int D) {
    int i = blockIdx.x * blockDim.x + threadIdx.x;
    if (i < tot) x[i] *= inv[i / D];
}

// ===========================================================================
// Host orchestration
// ===========================================================================
struct GvpW { const float *wh, *ws_b, *ws_w, *wv; };
struct LayerW {
    GvpW ff0, ff1, msg0, msg1, msg2;
    const float *n0_beta, *n0_gamma, *n1_beta, *n1_gamma;
};

extern "C" void kernel_launch(void* const* d_in, const int* in_sizes, int n_in,
                              void* d_out, int out_size, void* d_ws, size_t ws_size,
                              hipStream_t stream)
{
    (void)in_sizes; (void)out_size; (void)ws_size;
    const float* x    = (const float*)d_in[0];   // (NN,10)
    const float* xv   = (const float*)d_in[1];   // (NN,1,3)
    const int*   ei   = (const int*)d_in[2];     // (2,NE)
    const float* esc  = (const float*)d_in[3];   // (NE,1)
    const float* eav  = (const float*)d_in[4];   // (NE,1,3)
    const int*   bat  = (const int*)d_in[5];     // (NN,)
    const int* src = ei;
    const int* dst = ei + NE;

    // Params: jax pytree alphabetical leaf order. Supports both per-leaf
    // pointers (n_in > 7) and a single concatenated blob (n_in == 7).
    int pi = 6; size_t pofs = 0;
    const float* pblob = (n_in > 7) ? nullptr : (const float*)d_in[6];
    auto nf = [&](size_t n) -> const float* {
        if (pblob) { const float* p = pblob + pofs; pofs += n; return p; }
        return (const float*)d_in[pi++];
    };
    GvpW edge_in; edge_in.wh = nf(256); edge_in.ws_b = nf(256); edge_in.ws_w = nf(65792); edge_in.wv = nf(4096);
    LayerW L[4];
    for (int l = 0; l < 4; ++l) {
        L[l].ff0  = { nf(512),  nf(1024), nf(294912), nf(1024) };
        L[l].ff1  = { nf(1024), nf(256),  nf(270336), nf(512)  };
        L[l].msg0 = { nf(2304), nf(256),  nf(208896), nf(768)  };
        L[l].msg1 = { nf(256),  nf(256),  nf(69632),  nf(256)  };
        L[l].msg2 = { nf(256),  nf(256),  nf(69632),  nf(256)  };
        L[l].n0_beta = nf(256); L[l].n0_gamma = nf(256);
        L[l].n1_beta = nf(256); L[l].n1_gamma = nf(256);
    }
    GvpW node_in; node_in.wh = nf(16); node_in.ws_b = nf(256); node_in.ws_w = nf(6656); node_in.wv = nf(256);
    const float* nl_beta = nf(256); const float* nl_gamma = nf(256);
    GvpW proj; proj.wh = nf(256); proj.ws_b = nf(256); proj.ws_w = nf(69632); proj.wv = nf(16);

    // Workspace layout (~1.78 GB of fp32)
    float* w = (float*)d_ws;
    size_t off = 0;
    auto alloc = [&](size_t n) { float* p = w + off; off += n; return p; };
    float* s    = alloc((size_t)NN * 256);
    float* v    = alloc((size_t)NN * 48);
    float* es   = alloc((size_t)NE * 256);
    float* ev   = alloc((size_t)NE * 48);
    float* ms0  = alloc((size_t)NE * 256);   // also edge_in vn (E*256), project out
    float* ms1  = alloc((size_t)NE * 256);   // also ff0 output (NN*1024 == NE*256)
    float* vnb  = alloc((size_t)NE * 48);    // all vn buffers
    float* mv0  = alloc((size_t)NE * 48);
    float* mv1  = alloc((size_t)NE * 48);    // also ff0 vout (NN*96)
    float* ags  = alloc((size_t)NN * 256);
    float* agv  = alloc((size_t)NN * 48);
    float* cnt  = alloc(NN);
    float* gcnt = alloc(NG);
    float* P    = alloc(4096);
    float* outg = (float*)d_out;

    auto gemm = [&](const float* A0, int L0, int S0, const float* A1, int L1, int S1,
                    const float* A2, int L2, int S2, const float* A3, int L3, int S3,
                    const float* W, const float* b, float* C, int M, int K, int N, int act) {
        dim3 g((M + TM - 1) / TM, N / TN);
        gemm_gather_wmma<<<g, 256, 0, stream>>>(A0, L0, S0, A1, L1, S1, A2, L2, S2,
                                                A3, L3, S3, src, dst, W, b, C, M, K, N, act);
    };
    auto vnK = [&](const float* V0, int R0, int S0, const float* V1, int R1, int S1,
                   const float* V2, int R2, int S2, const float* wh, int H, float* out, int M) {
        int tot = M * H;
        gvp_vn_kernel<<<(tot + 255) / 256, 256, 0, stream>>>(V0, R0, S0, V1, R1, S1,
                                                             V2, R2, S2, src, dst, wh, H, out, tot);
    };
    auto voutK = [&](const float* V0, int R0, int S0, const float* V1, int R1, int S1,
                     const float* V2, int R2, int S2, int VO, float* out, int M, int act) {
        int tot = M * VO;
        gvp_vout_kernel<<<(tot + 255) / 256, 256, 0, stream>>>(V0, R0, S0, V1, R1, S1,
                                                               V2, R2, S2, src, dst, P, VO, out, tot, act);
    };
    auto pK = [&](const GvpW& g, int VI, int H, int VO) {
        small_matmul_kernel<<<(VI * VO + 63) / 64, 64, 0, stream>>>(g.wh, g.wv, P, VI, H, VO);
    };
    auto lnK = [&](float* ss, float* vv, const float* as, const float* av, const float* ic,
                   const float* ga, const float* be, int M) {
        ln_kernel<<<(M + 7) / 8, 256, 0, stream>>>(ss, vv, as, av, ic, ga, be, M);
    };

    // ---------------- node_in GVP + node_ln ----------------
    pK(node_in, 1, 16, 16);
    vnK(xv, 1, 0, 0, 0, 0, 0, 0, 0, node_in.wh, 16, vnb, NN);
    voutK(xv, 1, 0, 0, 0, 0, 0, 0, 0, 16, v, NN, 0);
    gemm(x, 10, 0, vnb, 16, 0, 0, 0, 0, 0, 0, 0, node_in.ws_w, node_in.ws_b, s, NN, 26, 256, 1);
    lnK(s, v, nullptr, nullptr, nullptr, nl_gamma, nl_beta, NN);

    // ---------------- edge_in GVP ----------------
    pK(edge_in, 1, 256, 16);
    vnK(eav, 1, 0, 0, 0, 0, 0, 0, 0, edge_in.wh, 256, ms0, NE);
    voutK(eav, 1, 0, 0, 0, 0, 0, 0, 0, 16, ev, NE, 1 /*sigmoid norm gate*/);
    gemm(esc, 1, 0, ms0, 256, 0, 0, 0, 0, 0, 0, 0, edge_in.ws_w, edge_in.ws_b, es, NE, 257, 256, 1);

    // ---------------- conv layers ----------------
    for (int l = 0; l < 4; ++l) {
        const LayerW& Lp = L[l];
        // msg0: Vin = [v[src] | ev | v[dst]]
        pK(Lp.msg0, 48, 48, 16);
        vnK(v, 16, 1, ev, 16, 0, v, 16, 2, Lp.msg0.wh, 48, vnb, NE);
        voutK(v, 16, 1, ev, 16, 0, v, 16, 2, 16, mv0, NE, 0);
        gemm(s, 256, 1, es, 256, 0, s, 256, 2, vnb, 48, 0,
             Lp.msg0.ws_w, Lp.msg0.ws_b, ms0, NE, 816, 256, 1);
        // msg1
        pK(Lp.msg1, 16, 16, 16);
        vnK(mv0, 16, 0, 0, 0, 0, 0, 0, 0, Lp.msg1.wh, 16, vnb, NE);
        voutK(mv0, 16, 0, 0, 0, 0, 0, 0, 0, 16, mv1, NE, 0);
        gemm(ms0, 256, 0, vnb, 16, 0, 0, 0, 0, 0, 0, 0,
             Lp.msg1.ws_w, Lp.msg1.ws_b, ms1, NE, 272, 256, 1);
        // msg2 (no act)
        pK(Lp.msg2, 16, 16, 16);
        vnK(mv1, 16, 0, 0, 0, 0, 0, 0, 0, Lp.msg2.wh, 16, vnb, NE);
        voutK(mv1, 16, 0, 0, 0, 0, 0, 0, 0, 16, mv0, NE, 0);
        gemm(ms1, 256, 0, vnb, 16, 0, 0, 0, 0, 0, 0, 0,
             Lp.msg2.ws_w, Lp.msg2.ws_b, ms0, NE, 272, 256, 0);
        // aggregation: segment-mean over dst
        hipMemsetAsync(ags, 0, (size_t)NN * 256 * 4, stream);
        hipMemsetAsync(agv, 0, (size_t)NN * 48 * 4, stream);
        hipMemsetAsync(cnt, 0, (size_t)NN * 4, stream);
        count_kernel<<<(NE + 255) / 256, 256, 0, stream>>>(dst, cnt, NE);
        scatter_add_kernel<<<(NE * 256 + 255) / 256, 256, 0, stream>>>(ms0, dst, ags, NE, 256);
        scatter_add_kernel<<<(NE * 48 + 255) / 256, 256, 0, stream>>>(mv0, dst, agv, NE, 48);
        inv_clamp_kernel<<<(NN + 255) / 256, 256, 0, stream>>>(cnt, NN);
        lnK(s, v, ags, agv, cnt, Lp.n0_gamma, Lp.n0_beta, NN);
        // ff0 (256,16 -> 1024,32)
        pK(Lp.ff0, 16, 32, 32);
        vnK(v, 16, 0, 0, 0, 0, 0, 0, 0, Lp.ff0.wh, 32, vnb, NN);
        voutK(v, 16, 0, 0, 0, 0, 0, 0, 0, 32, mv1, NN, 0);
        gemm(s, 256, 0, vnb, 32, 0, 0, 0, 0, 0, 0, 0,
             Lp.ff0.ws_w, Lp.ff0.ws_b, ms1, NN, 288, 1024, 1);
        // ff1 (1024,32 -> 256,16), no act
        pK(Lp.ff1, 32, 32, 16);
        vnK(mv1, 32, 0, 0, 0, 0, 0, 0, 0, Lp.ff1.wh, 32, vnb, NN);
        voutK(mv1, 32, 0, 0, 0, 0, 0, 0, 0, 16, mv0, NN, 0);
        gemm(ms1, 1024, 0, vnb, 32, 0, 0, 0, 0, 0, 0, 0,
             Lp.ff1.ws_w, Lp.ff1.ws_b, ms0, NN, 1056, 256, 0);
        lnK(s, v, ms0, mv0, nullptr, Lp.n1_gamma, Lp.n1_beta, NN);
    }

    // ---------------- project + graph mean pooling ----------------
    vnK(v, 16, 0, 0, 0, 0, 0, 0, 0, proj.wh, 16, vnb, NN);
    gemm(s, 256, 0, vnb, 16, 0, 0, 0, 0, 0, 0, 0,
         proj.ws_w, proj.ws_b, ms0, NN, 272, 256, 0);
    hipMemsetAsync(outg, 0, (size_t)NG * 256 * 4, stream);
    hipMemsetAsync(gcnt, 0, (size_t)NG * 4, stream);
    count_kernel<<<(NN + 255) / 256, 256, 0, stream>>>(bat, gcnt, NN);
    scatter_add_kernel<<<(NN * 256 + 255) / 256, 256, 0, stream>>>(ms0, bat, outg, NN, 256);
    inv_clamp_kernel<<<(NG + 255) / 256, 256, 0, stream>>>(gcnt, NG);
    scale_rows_kernel<<<(NG * 256 + 255) / 256, 256, 0, stream>>>(outg, gcnt, NG * 256, 256);
}